// SimpleNmsModule_86165633892928
// MI455X (gfx1250) — compile-verified
//
#include <hip/hip_runtime.h>

#define NBOX 8192
#define NTHR 1024   // 32 wave32 waves on one WGP
#define NSUPW (NBOX / 32)

// Monotone map: ascending unsigned <=> ascending float
static __device__ __forceinline__ unsigned fkey_asc(float f) {
  unsigned u = __float_as_uint(f);
  return (u & 0x80000000u) ? ~u : (u | 0x80000000u);
}

// -----------------------------------------------------------------------------
// Kernel 1: build 64-bit sort keys (descending score, ascending index tiebreak),
// bitonic sort in LDS, then gather boxes into SoA workspace arrays + areas.
// -----------------------------------------------------------------------------
__global__ __launch_bounds__(NTHR, 1)
void nms_sort_gather(const float* __restrict__ boxes,
                     const float* __restrict__ scores,
                     int* __restrict__ sorted_idx,
                     float* __restrict__ sx1, float* __restrict__ sy1,
                     float* __restrict__ sx2, float* __restrict__ sy2,
                     float* __restrict__ sar) {
  __shared__ unsigned long long sk[NBOX];   // 64 KB
  const int t = threadIdx.x;

  // CDNA5 prefetch path (global_prefetch_b8) for the gather source
  __builtin_prefetch(boxes + t * 4, 0, 0);

  for (int i = t; i < NBOX; i += NTHR) {
    unsigned u = fkey_asc(scores[i]);
    // ~u: descending score; low 32: ascending original index (stable tiebreak)
    sk[i] = (((unsigned long long)(~u)) << 32) | (unsigned)i;
  }
  __syncthreads();

  // Bitonic sort, ascending on the packed key
  for (int k = 2; k <= NBOX; k <<= 1) {
    for (int j = k >> 1; j > 0; j >>= 1) {
      for (int m = t; m < NBOX; m += NTHR) {
        int x = m ^ j;
        if (x > m) {
          unsigned long long a = sk[m], b = sk[x];
          bool asc = ((m & k) == 0);
          if ((a > b) == asc) { sk[m] = b; sk[x] = a; }
        }
      }
      __syncthreads();
    }
  }

  const float4* b4 = (const float4*)boxes;
  for (int i = t; i < NBOX; i += NTHR) {
    int idx = (int)(unsigned)(sk[i] & 0xffffffffull);
    sorted_idx[i] = idx;
    float4 bb = b4[idx];                       // global_load_b128 gather
    sx1[i] = bb.x; sy1[i] = bb.y; sx2[i] = bb.z; sy2[i] = bb.w;
    sar[i] = (bb.z - bb.x) * (bb.w - bb.y);
  }
}

// -----------------------------------------------------------------------------
// Async global -> LDS staging (CDNA5 async path, ASYNCcnt-tracked).
// The low 32 bits of a generic shared pointer are the wave-relative LDS offset
// (shared aperture keeps the offset in addr[31:0]), which is exactly what the
// VDST operand of global_load_async_to_lds_* consumes.
// -----------------------------------------------------------------------------
__device__ __forceinline__ void async_copy_b128(const float* __restrict__ g,
                                                float* l, int n, int t) {
  for (int i = t * 4; i < n; i += NTHR * 4) {
    unsigned loff   = (unsigned)(size_t)(l + i);
    const void* gp  = (const void*)(g + i);
    asm volatile("global_load_async_to_lds_b128 %0, %1, off"
                 :: "v"(loff), "v"(gp) : "memory");
  }
}

// -----------------------------------------------------------------------------
// Kernel 2: single-workgroup greedy NMS entirely out of LDS.
// Word-wise bitmask scan: one LDS read per 32 candidates; suppressed boxes cost
// only a bit test. Early (uniform) exit once max_outputs boxes are kept.
// Dynamic LDS layout: sx1|sy1|sx2|sy2|sar (5 * 32KB) + suppression bitmask 1KB.
// -----------------------------------------------------------------------------
__global__ __launch_bounds__(NTHR, 1)
void nms_sequential(const float* __restrict__ gx1, const float* __restrict__ gy1,
                    const float* __restrict__ gx2, const float* __restrict__ gy2,
                    const float* __restrict__ gar,
                    const int* __restrict__ sorted_idx,
                    const float* __restrict__ thr_p,
                    const int* __restrict__ maxo_p,
                    int* __restrict__ out, int out_cap) {
  extern __shared__ float smemf[];
  float* sx1 = smemf;
  float* sy1 = sx1 + NBOX;
  float* sx2 = sy1 + NBOX;
  float* sy2 = sx2 + NBOX;
  float* sar = sy2 + NBOX;
  unsigned* sup = (unsigned*)(sar + NBOX);   // NSUPW = 256 words

  const int t = threadIdx.x;

  async_copy_b128(gx1, sx1, NBOX, t);
  async_copy_b128(gy1, sy1, NBOX, t);
  async_copy_b128(gx2, sx2, NBOX, t);
  async_copy_b128(gy2, sy2, NBOX, t);
  async_copy_b128(gar, sar, NBOX, t);
  for (int i = t; i < NSUPW; i += NTHR) sup[i] = 0u;

  asm volatile("s_wait_asynccnt 0" ::: "memory");   // drain ASYNCcnt
  __syncthreads();

  const float thr = *thr_p;
  int mo = *maxo_p;
  if (mo > out_cap) mo = out_cap;

  int kept = 0;
  bool done = (mo <= 0);
  for (int w = 0; w < NSUPW && !done; ++w) {
    unsigned cur = sup[w];                       // one LDS read per 32 boxes
    for (int b = 0; b < 32; ++b) {
      if ((cur >> b) & 1u) continue;             // suppressed: bit test only
      const int i = (w << 5) | b;
      if (t == 0) out[kept] = sorted_idx[i];
      ++kept;
      if (kept >= mo) { done = true; break; }    // uniform early exit

      const float xx1 = sx1[i], yy1 = sy1[i];
      const float xx2 = sx2[i], yy2 = sy2[i];
      const float ai  = sar[i];
      for (int j = i + 1 + t; j < NBOX; j += NTHR) {
        float iw = fminf(xx2, sx2[j]) - fmaxf(xx1, sx1[j]);
        float ih = fminf(yy2, sy2[j]) - fmaxf(yy1, sy1[j]);
        iw = fmaxf(iw, 0.0f);
        ih = fmaxf(ih, 0.0f);
        float inter = iw * ih;
        float iou = inter / (ai + sar[j] - inter);
        if (iou > thr) atomicOr(&sup[j >> 5], 1u << (j & 31));  // ds_or_b32
      }
      __syncthreads();
      cur = sup[w];                              // re-read word updated by this step
    }
  }
  __syncthreads();
  for (int k2 = kept + t; k2 < mo; k2 += NTHR) out[k2] = -1;   // pad
}

// -----------------------------------------------------------------------------
// Host launcher. Inputs: boxes[8192*4] f32, scores[8192] f32,
// iou_threshold (f32 scalar), max_outputs (i32 scalar). Output: 100 x i32.
// Workspace: sorted_idx + 5 SoA arrays = 192 KB.
// -----------------------------------------------------------------------------
extern "C" void kernel_launch(void* const* d_in, const int* in_sizes, int n_in,
                              void* d_out, int out_size, void* d_ws, size_t ws_size,
                              hipStream_t stream) {
  (void)in_sizes; (void)n_in; (void)ws_size;
  const float* boxes  = (const float*)d_in[0];
  const float* scores = (const float*)d_in[1];
  const float* thr    = (const float*)d_in[2];
  const int*   maxo   = (const int*)d_in[3];
  int* out = (int*)d_out;

  char* ws = (char*)d_ws;
  int*   sorted_idx = (int*)(ws);
  float* sx1 = (float*)(ws + 1 * NBOX * 4);
  float* sy1 = (float*)(ws + 2 * NBOX * 4);
  float* sx2 = (float*)(ws + 3 * NBOX * 4);
  float* sy2 = (float*)(ws + 4 * NBOX * 4);
  float* sar = (float*)(ws + 5 * NBOX * 4);

  nms_sort_gather<<<1, NTHR, 0, stream>>>(boxes, scores, sorted_idx,
                                          sx1, sy1, sx2, sy2, sar);

  size_t shmem = (size_t)(5 * NBOX * 4) + (size_t)NSUPW * 4;  // ~164 KB < 320 KB/WGP
  nms_sequential<<<1, NTHR, shmem, stream>>>(sx1, sy1, sx2, sy2, sar,
                                             sorted_idx, thr, maxo, out, out_size);
}